// CostMapLayer_11888469476362
// MI455X (gfx1250) — compile-verified
//
#include <hip/hip_runtime.h>

// ---------------------------------------------------------------------------
// CostMapLayer for MI455X (gfx1250, wave32).
//
// B=8, N=500000 points -> scatter-min + count into an 8x512x512 grid.
// Strategy: grid (16 MB) lives in the 192 MB L2; streaming inputs/outputs use
// non-temporal cache hints so they don't evict it. Scatter uses no-return
// global atomics (global_atomic_min_u32 / global_atomic_add_u32). Float min
// is done on a monotonic uint encoding of the float -> bit-exact min, no
// dependence on f32 atomic-min semantics.
// d_out layout: [0, CELLS) = cost (f32), [CELLS, 2*CELLS) = mask (int32).
// d_out's two halves are also the atomic scratch (keys / counts) -> no d_ws
// needed, and every launch re-initializes them (deterministic under graph
// replay).
// ---------------------------------------------------------------------------

#define BATCH   8
#define NPTS    500000
#define HGT     512
#define WID     512
#define CELLS   (BATCH * HGT * WID)     // 2,097,152
#define PAIRS   (BATCH * NPTS / 2)      // 2,000,000 (two points per thread)
#define PAIRS_PER_BATCH (NPTS / 2)      // 250,000

typedef float    v4f __attribute__((ext_vector_type(4)));
typedef float    v2f __attribute__((ext_vector_type(2)));
typedef unsigned v4u __attribute__((ext_vector_type(4)));
typedef int      v4i __attribute__((ext_vector_type(4)));

// Monotonic uint encoding: preserves float ordering under unsigned compare.
__device__ __forceinline__ unsigned fkey(float f) {
    unsigned u = __float_as_uint(f);
    return (u & 0x80000000u) ? ~u : (u ^ 0x80000000u);
}
__device__ __forceinline__ float funkey(unsigned k) {
    unsigned u = (k & 0x80000000u) ? (k ^ 0x80000000u) : ~k;
    return __uint_as_float(u);
}

// ---------------------------------------------------------------------------
// Kernel 1: initialize key grid to +inf-key, count grid to 0. b128 stores.
// One thread per 4 cells: CELLS/4 = 524288 threads.
// ---------------------------------------------------------------------------
__global__ void __launch_bounds__(256)
costmap_init(v4u* __restrict__ keys4, v4u* __restrict__ cnts4) {
    const int i = blockIdx.x * blockDim.x + threadIdx.x;
    if (i < CELLS / 4) {
        v4u ff = {0xFFFFFFFFu, 0xFFFFFFFFu, 0xFFFFFFFFu, 0xFFFFFFFFu};
        v4u zz = {0u, 0u, 0u, 0u};
        keys4[i] = ff;   // stay regular-temporal: this data is re-read by atomics
        cnts4[i] = zz;
    }
}

// ---------------------------------------------------------------------------
// Kernel 2: scatter. One thread = one *pair* of consecutive points of the
// same batch (N even): global_load_b128 for coords, b64 for costs, both with
// non-temporal hint (streamed once). Atomics are no-return -> STOREcnt path.
// ---------------------------------------------------------------------------
__device__ __forceinline__ void scatter_one(float x, float y, float c, int b,
                                            unsigned* __restrict__ keys,
                                            unsigned* __restrict__ cnts) {
    // reference: ix = floor(x + 0.5) as int32 (HALF_CENTOR=True)
    const int ix = (int)floorf(x + 0.5f);
    const int iy = (int)floorf(y + 0.5f);
    if ((unsigned)ix < (unsigned)WID && (unsigned)iy < (unsigned)HGT) {
        const int seg = (b * HGT + iy) * WID + ix;
        atomicMin(&keys[seg], fkey(c));   // result unused -> no-return atomic
        atomicAdd(&cnts[seg], 1u);
    }
}

__global__ void __launch_bounds__(256)
costmap_scatter(const v4f* __restrict__ pts4,   // [PAIRS] = 2 points (x,y,x,y)
                const v2f* __restrict__ cst2,   // [PAIRS] = 2 costs
                unsigned* __restrict__ keys,
                unsigned* __restrict__ cnts) {
    const int p = blockIdx.x * blockDim.x + threadIdx.x;
    if (p >= PAIRS) return;

    const v4f pt = __builtin_nontemporal_load(&pts4[p]);  // th:NT
    const v2f cc = __builtin_nontemporal_load(&cst2[p]);  // th:NT
    const int b  = p / PAIRS_PER_BATCH;                   // magic-mul, no div

    scatter_one(pt.x, pt.y, cc.x, b, keys, cnts);
    scatter_one(pt.z, pt.w, cc.y, b, keys, cnts);
}

// ---------------------------------------------------------------------------
// Kernel 3: finalize in place. One thread per 4 cells; b128 NT stores
// (write-once output, keep it out of the way of the atomics still landing
// elsewhere in L2 on other replays).
// ---------------------------------------------------------------------------
__global__ void __launch_bounds__(256)
costmap_finalize(unsigned* __restrict__ buf,          // d_out as u32[2*CELLS]
                 const float* __restrict__ defc_ptr) {
    const int i = blockIdx.x * blockDim.x + threadIdx.x;
    if (i >= CELLS / 4) return;

    const float defc = *defc_ptr;   // uniform -> s_load + broadcast

    const v4u k = ((const v4u*)buf)[i];
    const v4u c = ((const v4u*)(buf + CELLS))[i];

    v4f cost;
    v4i mask;
#pragma unroll
    for (int j = 0; j < 4; ++j) {
        const unsigned cnt = c[j];
        cost[j] = (cnt != 0u) ? funkey(k[j]) : defc;
        mask[j] = (int)cnt - 1;     // -1 for empty cells (MIN semantics)
    }

    __builtin_nontemporal_store(cost, (v4f*)buf + i);               // th:NT
    __builtin_nontemporal_store(mask, (v4i*)(buf + CELLS) + i);     // th:NT
}

// ---------------------------------------------------------------------------
// Launch: init (2048 blk) -> scatter (7813 blk) -> finalize (2048 blk),
// serialized on `stream`. 256 threads/block = 8 wave32 waves.
// ---------------------------------------------------------------------------
extern "C" void kernel_launch(void* const* d_in, const int* in_sizes, int n_in,
                              void* d_out, int out_size, void* d_ws, size_t ws_size,
                              hipStream_t stream) {
    (void)in_sizes; (void)n_in; (void)out_size; (void)d_ws; (void)ws_size;

    const v4f*   pts4 = (const v4f*)d_in[0];    // points  [B,N,2] f32
    const v2f*   cst2 = (const v2f*)d_in[1];    // costs   [B,N]   f32
    const float* defc = (const float*)d_in[2];  // default_cost scalar (device)
    // d_in[3]/d_in[4] = height/width (compile-time constants 512 here)

    unsigned* keys = (unsigned*)d_out;          // first  CELLS u32 -> cost
    unsigned* cnts = keys + CELLS;              // second CELLS u32 -> mask

    constexpr int TB = 256;
    costmap_init    <<<(CELLS / 4 + TB - 1) / TB, TB, 0, stream>>>(
        (v4u*)keys, (v4u*)cnts);
    costmap_scatter <<<(PAIRS + TB - 1) / TB, TB, 0, stream>>>(
        pts4, cst2, keys, cnts);
    costmap_finalize<<<(CELLS / 4 + TB - 1) / TB, TB, 0, stream>>>(
        (unsigned*)d_out, defc);
}